// SetAbstraction_29978871726618
// MI455X (gfx1250) — compile-verified
//
#include <hip/hip_runtime.h>
#include <hip/hip_bf16.h>

typedef float v2f __attribute__((ext_vector_type(2)));
typedef float v8f __attribute__((ext_vector_type(8)));

// Problem constants
#define BATCH   16
#define NPTS    4096
#define NPOINT  1024
#define NSAMPLE 32
#define NTOT    524288L      // BATCH*NPOINT*NSAMPLE
#define BPTOT   16384        // BATCH*NPOINT
#define NBLK    1024         // conv grid.x (N partitions)
#define CONV_NITER 32        // 16-position tiles per block -> 512 positions/block

// ---------------------------------------------------------------------------
// K1: brute-force 32-NN per query + write new_xyz to d_out[0 .. 49151]
// One thread per query; insertion-sorted top-32 lists live in padded LDS.
// ---------------------------------------------------------------------------
#define KNN_BLOCK 128
#define KNN_CHUNK 512

__global__ __launch_bounds__(KNN_BLOCK)
void knn_kernel(const float* __restrict__ xyz, const int* __restrict__ sample_idx,
                int* __restrict__ idx_out, float* __restrict__ out)
{
    __shared__ float sx[KNN_CHUNK], sy[KNN_CHUNK], sz[KNN_CHUNK];
    __shared__ float sd[KNN_BLOCK * 33];
    __shared__ int   si[KNN_BLOCK * 33];

    const int b = blockIdx.y;
    const int p = blockIdx.x * KNN_BLOCK + threadIdx.x;
    const float* xb = xyz + (long)b * 3 * NPTS;

    const int qi = sample_idx[p];
    const float qx = xb[qi], qy = xb[NPTS + qi], qz = xb[2 * NPTS + qi];

    // new_xyz straight to output (shape B x 3 x NPOINT)
    out[(b * 3 + 0) * NPOINT + p] = qx;
    out[(b * 3 + 1) * NPOINT + p] = qy;
    out[(b * 3 + 2) * NPOINT + p] = qz;

    float* dl = sd + threadIdx.x * 33;
    int*   il = si + threadIdx.x * 33;
    for (int j = 0; j < NSAMPLE; ++j) { dl[j] = 3.4e38f; il[j] = 0; }
    float worst = 3.4e38f;

    for (int n0 = 0; n0 < NPTS; n0 += KNN_CHUNK) {
        __syncthreads();
        for (int t = threadIdx.x; t < KNN_CHUNK; t += KNN_BLOCK) {
            sx[t] = xb[n0 + t];
            sy[t] = xb[NPTS + n0 + t];
            sz[t] = xb[2 * NPTS + n0 + t];
        }
        __syncthreads();
        for (int t = 0; t < KNN_CHUNK; ++t) {
            const float dx = sx[t] - qx, dy = sy[t] - qy, dz = sz[t] - qz;
            const float d = dx * dx + dy * dy + dz * dz;
            if (d < worst) {               // strict < keeps earlier index on ties (matches top_k)
                int j = NSAMPLE - 1;
                while (j > 0 && dl[j - 1] > d) {   // stable insertion (after equals)
                    dl[j] = dl[j - 1]; il[j] = il[j - 1]; --j;
                }
                dl[j] = d; il[j] = n0 + t;
                worst = dl[NSAMPLE - 1];
            }
        }
    }
    int* op = idx_out + ((long)(b * NPOINT + p)) * NSAMPLE;
    for (int j = 0; j < NSAMPLE; ++j) op[j] = il[j];
}

// ---------------------------------------------------------------------------
// Conv + stats kernel (templated over layer):
//  MODE 0: gather 67-ch feature, W1 (64x67) -> h1pre + stat partials
//  MODE 1: BN1-affine+relu(h1pre), W2 (64x64) -> h2pre + stat partials
//  MODE 2: BN2-affine+relu(h2pre), W3 (128x64) -> per-(o,bp) max/min + partials
// Block: 128 threads (4 waves); wave w handles M-tile (blockIdx.y*4+w)*16.
// Each wave: D(16x16) += A(16x4, weights) x B(4x16, features) via
// v_wmma_f32_16x16x4_f32, K/4 steps per 16-position tile, 32 tiles per block.
// ---------------------------------------------------------------------------
template <int MODE>
__global__ __launch_bounds__(128)
void conv_kernel(const float* __restrict__ src,     // xyz (MODE0) or h_prev
                 const float* __restrict__ points,  // MODE0 only
                 const int*   __restrict__ gidx,    // MODE0 only
                 const float* __restrict__ newxyz,  // MODE0 only (d_out region)
                 const float* __restrict__ W,
                 const float* __restrict__ bias,
                 const float* __restrict__ aPrev,   // folded BN scale of prev layer
                 const float* __restrict__ cPrev,   // folded BN shift of prev layer
                 float* __restrict__ hout,          // null for MODE2
                 float* __restrict__ psum,          // [Cout][NBLK] partials
                 float* __restrict__ psq,
                 float* __restrict__ mxbuf,         // MODE2: [128][BPTOT]
                 float* __restrict__ mnbuf)
{
    constexpr int Cin    = (MODE == 0) ? 67 : 64;
    constexpr int KP     = (MODE == 0) ? 68 : 64;   // K padded to multiple of 4
    constexpr int KSTEPS = KP / 4;

    __shared__ float FT[KP * 16];
    __shared__ float sA[64], sC[64];
    __shared__ int   sIdx[16];

    const int tid   = threadIdx.x;
    const int lane  = tid & 31;
    const int wv    = tid >> 5;
    const int nl    = lane & 15;
    const int hi    = lane >> 4;          // 0: lanes 0-15, 1: lanes 16-31
    const int khalf = hi * 2;             // A/B K-offset for upper half-wave
    const int mbase = (blockIdx.y * 4 + wv) * 16;
    const int m     = mbase + nl;

    // A-fragments: W tile rows, ISA layout (v0: K=0|2, v1: K=1|3 per half-wave)
    v2f afrag[KSTEPS];
    for (int ks = 0; ks < KSTEPS; ++ks) {
        const int k0 = ks * 4 + khalf;
        afrag[ks].x = (k0     < Cin) ? W[m * Cin + k0]     : 0.f;
        afrag[ks].y = (k0 + 1 < Cin) ? W[m * Cin + k0 + 1] : 0.f;
    }
    // bias per C/D row (cancels in BN but kept for fidelity)
    float bv[8];
    for (int i = 0; i < 8; ++i) bv[i] = bias[mbase + i + hi * 8];

    if (MODE != 0) {
        for (int c = tid; c < 64; c += 128) { sA[c] = aPrev[c]; sC[c] = cPrev[c]; }
    }

    float ssum[8], ssq[8];
    for (int i = 0; i < 8; ++i) { ssum[i] = 0.f; ssq[i] = 0.f; }
    float pmx[8], pmn[8];
    if (MODE == 2) for (int i = 0; i < 8; ++i) { pmx[i] = -3.4e38f; pmn[i] = 3.4e38f; }

    const long nBase = (long)blockIdx.x * (16 * CONV_NITER);

    for (int t = 0; t < CONV_NITER; ++t) {
        const long n0 = nBase + (long)t * 16;
        __syncthreads();
        if (MODE == 0) {
            if (tid < 16) sIdx[tid] = gidx[n0 + tid];
        }
        __syncthreads();

        // Stage K x 16 feature tile into LDS
        for (int e = tid; e < KP * 16; e += 128) {
            const int c = e >> 4, j = e & 15;
            const long n = n0 + j;
            float val;
            if (MODE == 0) {
                const int bp = (int)(n >> 5);
                const int b = bp >> 10, p = bp & 1023;
                const int xi = sIdx[j];
                if (c < 3)
                    val = src[((long)b * 3 + c) * NPTS + xi] - newxyz[(b * 3 + c) * NPOINT + p];
                else if (c < 67)
                    val = points[((long)b * 64 + (c - 3)) * NPTS + xi];
                else
                    val = 0.f;
            } else {
                const float h = src[(long)c * NTOT + n];
                val = fmaxf(fmaf(h, sA[c], sC[c]), 0.f);   // BN-affine + ReLU on the fly
            }
            FT[e] = val;
        }
        __syncthreads();

        // Accumulate D = sum_k A_k x B_k  (+ bias) with fp32 WMMA
        v8f acc;
        for (int i = 0; i < 8; ++i) acc[i] = bv[i];
        for (int ks = 0; ks < KSTEPS; ++ks) {
            const int k = ks * 4 + khalf;
            v2f bf;
            bf.x = FT[k * 16 + nl];
            bf.y = FT[(k + 1) * 16 + nl];
            acc = __builtin_amdgcn_wmma_f32_16x16x4_f32(
                false, afrag[ks], false, bf, (short)0, acc, false, false);
        }

        // Consume tile: stats always; h store for MODE0/1; max/min for MODE2
        const long ncol = n0 + nl;
        for (int i = 0; i < 8; ++i) {
            const float v = acc[i];
            ssum[i] += v;
            ssq[i]  += v * v;
            if (MODE != 2) {
                const int o = mbase + i + hi * 8;
                hout[(long)o * NTOT + ncol] = v;
            }
        }
        if (MODE == 2) {
            for (int i = 0; i < 8; ++i) {
                float rmx = acc[i], rmn = acc[i];
                for (int msk = 1; msk < 16; msk <<= 1) {
                    rmx = fmaxf(rmx, __shfl_xor(rmx, msk, 32));
                    rmn = fminf(rmn, __shfl_xor(rmn, msk, 32));
                }
                pmx[i] = fmaxf(pmx[i], rmx);
                pmn[i] = fminf(pmn[i], rmn);
            }
            if (t & 1) {   // two 16-wide tiles complete one 32-sample group
                const int bp = (int)(n0 >> 5);
                if (nl == 0) {
                    for (int i = 0; i < 8; ++i) {
                        const int o = mbase + i + hi * 8;
                        mxbuf[(long)o * BPTOT + bp] = pmx[i];
                        mnbuf[(long)o * BPTOT + bp] = pmn[i];
                    }
                }
                for (int i = 0; i < 8; ++i) { pmx[i] = -3.4e38f; pmn[i] = 3.4e38f; }
            }
        }
    }

    // Deterministic per-block stat partials (no atomics)
    for (int i = 0; i < 8; ++i) {
        float a = ssum[i], q = ssq[i];
        for (int msk = 1; msk < 16; msk <<= 1) {
            a += __shfl_xor(a, msk, 32);
            q += __shfl_xor(q, msk, 32);
        }
        if (nl == 0) {
            const int o = mbase + i + hi * 8;
            psum[(long)o * NBLK + blockIdx.x] = a;
            psq [(long)o * NBLK + blockIdx.x] = q;
        }
    }
}

// ---------------------------------------------------------------------------
// Reduce per-block partials -> mean/var -> folded BN affine (a, c)
// ---------------------------------------------------------------------------
__global__ __launch_bounds__(256)
void finalize_stats(const float* __restrict__ psum, const float* __restrict__ psq,
                    const float* __restrict__ g, const float* __restrict__ be,
                    float* __restrict__ aOut, float* __restrict__ cOut)
{
    const int o = blockIdx.x;
    __shared__ float s1[256], s2[256];
    float a = 0.f, q = 0.f;
    for (int i = threadIdx.x; i < NBLK; i += 256) {
        a += psum[(long)o * NBLK + i];
        q += psq [(long)o * NBLK + i];
    }
    s1[threadIdx.x] = a; s2[threadIdx.x] = q;
    __syncthreads();
    for (int st = 128; st > 0; st >>= 1) {
        if (threadIdx.x < st) {
            s1[threadIdx.x] += s1[threadIdx.x + st];
            s2[threadIdx.x] += s2[threadIdx.x + st];
        }
        __syncthreads();
    }
    if (threadIdx.x == 0) {
        const float inv  = 1.f / (float)NTOT;
        const float mean = s1[0] * inv;
        const float var  = s2[0] * inv - mean * mean;
        const float r    = rsqrtf(var + 1e-5f);
        const float av   = g[o] * r;
        aOut[o] = av;
        cOut[o] = be[o] - mean * av;
    }
}

// ---------------------------------------------------------------------------
// max_s BN3(h3) = a*max(h3)+c if a>=0 else a*min(h3)+c
// ---------------------------------------------------------------------------
__global__ __launch_bounds__(256)
void final_max_kernel(const float* __restrict__ mx, const float* __restrict__ mn,
                      const float* __restrict__ a3, const float* __restrict__ c3,
                      float* __restrict__ out)
{
    const long t = (long)blockIdx.x * 256 + threadIdx.x;   // [0, 128*BPTOT)
    const int o  = (int)(t >> 14);
    const int bp = (int)(t & (BPTOT - 1));
    const int b = bp >> 10, p = bp & 1023;
    const float a = a3[o], c = c3[o];
    const float v = (a >= 0.f) ? mx[t] : mn[t];
    out[(long)BATCH * 3 * NPOINT + ((long)(b * 128 + o)) * NPOINT + p] = fmaf(a, v, c);
}

// ---------------------------------------------------------------------------
extern "C" void kernel_launch(void* const* d_in, const int* in_sizes, int n_in,
                              void* d_out, int out_size, void* d_ws, size_t ws_size,
                              hipStream_t stream)
{
    const float* xyz        = (const float*)d_in[0];
    const float* points     = (const float*)d_in[1];
    const int*   sample_idx = (const int*)  d_in[2];
    const float* W1 = (const float*)d_in[3];  const float* b1 = (const float*)d_in[4];
    const float* g1 = (const float*)d_in[5];  const float* be1 = (const float*)d_in[6];
    const float* W2 = (const float*)d_in[7];  const float* b2 = (const float*)d_in[8];
    const float* g2 = (const float*)d_in[9];  const float* be2 = (const float*)d_in[10];
    const float* W3 = (const float*)d_in[11]; const float* b3 = (const float*)d_in[12];
    const float* g3 = (const float*)d_in[13]; const float* be3 = (const float*)d_in[14];

    float* out = (float*)d_out;
    float* ws  = (float*)d_ws;
    int*   idx = (int*)d_ws;

    long off = NTOT;                         // idx occupies NTOT ints
    float* H1  = ws + off; off += 64L  * NTOT;
    float* H2  = ws + off; off += 64L  * NTOT;
    float* MX  = ws + off; off += 128L * BPTOT;
    float* MN  = ws + off; off += 128L * BPTOT;
    float* P1S = ws + off; off += 64L  * NBLK;
    float* P1Q = ws + off; off += 64L  * NBLK;
    float* P2S = ws + off; off += 64L  * NBLK;
    float* P2Q = ws + off; off += 64L  * NBLK;
    float* P3S = ws + off; off += 128L * NBLK;
    float* P3Q = ws + off; off += 128L * NBLK;
    float* A1  = ws + off; off += 64;
    float* C1  = ws + off; off += 64;
    float* A2  = ws + off; off += 64;
    float* C2  = ws + off; off += 64;
    float* A3  = ws + off; off += 128;
    float* C3  = ws + off; off += 128;

    // K1: knn + new_xyz
    knn_kernel<<<dim3(NPOINT / KNN_BLOCK, BATCH), KNN_BLOCK, 0, stream>>>(
        xyz, sample_idx, idx, out);

    // Layer 1: gather + GEMM (WMMA f32) + stat partials
    conv_kernel<0><<<dim3(NBLK, 1), 128, 0, stream>>>(
        xyz, points, idx, out, W1, b1, nullptr, nullptr, H1, P1S, P1Q, nullptr, nullptr);
    finalize_stats<<<64, 256, 0, stream>>>(P1S, P1Q, g1, be1, A1, C1);

    // Layer 2
    conv_kernel<1><<<dim3(NBLK, 1), 128, 0, stream>>>(
        H1, nullptr, nullptr, nullptr, W2, b2, A1, C1, H2, P2S, P2Q, nullptr, nullptr);
    finalize_stats<<<64, 256, 0, stream>>>(P2S, P2Q, g2, be2, A2, C2);

    // Layer 3: no h3 materialization; per-(o,bp) max/min + partials
    conv_kernel<2><<<dim3(NBLK, 2), 128, 0, stream>>>(
        H2, nullptr, nullptr, nullptr, W3, b3, A2, C2, nullptr, P3S, P3Q, MX, MN);
    finalize_stats<<<128, 256, 0, stream>>>(P3S, P3Q, g3, be3, A3, C3);

    // Output new_points
    final_max_kernel<<<(128 * BPTOT) / 256, 256, 0, stream>>>(MX, MN, A3, C3, out);
}